// Lintra_89000312307761
// MI455X (gfx1250) — compile-verified
//
#include <hip/hip_runtime.h>

// ---------------------------------------------------------------------------
// Lintra loss for MI455X (gfx1250).
// Phase 1: double-buffered TDM-staged segment scatter-reduction
//          (memory-bound: ~1.09 GB mandatory reads -> ~46us floor @ 23.3 TB/s)
// Phase 2: per-(image,class) pairwise L1 of instance means + smooth-L1
// Phase 3: final scalar reduction
// ---------------------------------------------------------------------------

#define N_CLASSES 5
#define IMAXV     32
#define NSEG      (N_CLASSES * (IMAXV + 1))   // 165
#define DIM       128
#define BDIM      8
#define HW        (512 * 512)
#define TILE      256                          // pixels per tile == blockDim
#define DQ        32                           // channels per TDM chunk (32 KB)
#define NCHUNK    (DIM / DQ)                   // 4 chunks, 2 stage buffers
#define LPAD      129                          // pad: (seg*129+d) spreads LDS banks
#define PPAD      133                          // phase-2 mean pad (133 mod 64 = 5)
#define BLOCKS_PER_IMG 64

#if defined(__has_builtin)
#if __has_builtin(__builtin_amdgcn_tensor_load_to_lds) && \
    __has_builtin(__builtin_amdgcn_s_wait_tensorcnt)
#define USE_TDM 1
#endif
#endif

#ifdef USE_TDM
typedef unsigned int u32x4 __attribute__((ext_vector_type(4)));
typedef int          i32x8 __attribute__((ext_vector_type(8)));
typedef int          i32x4 __attribute__((ext_vector_type(4)));

// Issue one TDM 2D tile load: X = tile_x contiguous f32 elements,
// Y = tile_y rows with row stride = row_stride_elems (in elements).
__device__ __forceinline__ void tdm_load_2d(unsigned lds_byte_addr,
                                            const void* gptr,
                                            unsigned tile_x, unsigned tile_y,
                                            unsigned tensor_x,
                                            unsigned row_stride_elems) {
  const unsigned long long ga = (unsigned long long)gptr;
  u32x4 g0;
  g0.x = 1u;                                          // count=1 (valid user D#)
  g0.y = lds_byte_addr;                               // lds_addr
  g0.z = (unsigned)ga;                                // global_addr[31:0]
  g0.w = (unsigned)((ga >> 32) & 0x1FFFFFFu)          // global_addr[56:32]
         | (2u << 30);                                // type = 2 ("image")
  const unsigned td0 = tensor_x;                      // tensor_dim0
  const unsigned td1 = tile_y;                        // tensor_dim1
  i32x8 g1;
  g1[0] = (int)(2u << 16);                            // data_size = 4B; mask=0
  g1[1] = (int)((td0 & 0xFFFFu) << 16);               // tensor_dim0[15:0]
  g1[2] = (int)((td0 >> 16) | ((td1 & 0xFFFFu) << 16));
  g1[3] = (int)((td1 >> 16) | (tile_x << 16));        // tile_dim0
  g1[4] = (int)(tile_y & 0xFFFFu);                    // tile_dim1 (tile_dim2=0)
  g1[5] = (int)row_stride_elems;                      // tensor_dim0_stride[31:0]
  g1[6] = 0;                                          // stride hi / dim1_stride
  g1[7] = 0;
  const i32x4 z4 = {0, 0, 0, 0};                      // groups 2/3 unused (2D)
  const i32x8 z8 = {0, 0, 0, 0, 0, 0, 0, 0};          // clang-23 6-arg form
  __builtin_amdgcn_tensor_load_to_lds(g0, g1, z4, z4, z8, 0);
}
#endif

// ---------------------------------------------------------------------------
// Phase 0: zero workspace (sums, counts, scalar accumulators)
// ---------------------------------------------------------------------------
__global__ void zero_ws_kernel(float* __restrict__ p, int n) {
  const int i = blockIdx.x * blockDim.x + threadIdx.x;
  if (i < n) p[i] = 0.0f;
}

// ---------------------------------------------------------------------------
// Phase 1: segment sums + counts.
// grid = (BLOCKS_PER_IMG, B), block = 256 (8 x wave32)
// LDS: 85,140 (accum) + 660 (counts) + 2*32,768 (stages) ~= 151 KB
//      -> 2 workgroups per 320 KB WGP.
// ---------------------------------------------------------------------------
__global__ __launch_bounds__(TILE) void seg_reduce_kernel(
    const float* __restrict__ feat, const int* __restrict__ labels,
    const int* __restrict__ indexes, float* __restrict__ gsum,
    float* __restrict__ gcount) {
  __shared__ float lsum[NSEG * LPAD];
  __shared__ float lcount[NSEG];
#ifdef USE_TDM
  __shared__ float stage[2][DQ * TILE];  // double-buffered TDM landing zone
#endif

  const int b = blockIdx.y;
  const int tid = threadIdx.x;

  for (int i = tid; i < NSEG * LPAD; i += TILE) lsum[i] = 0.0f;
  if (tid < NSEG) lcount[tid] = 0.0f;
  __syncthreads();

  const float* fb = feat + (size_t)b * DIM * HW;
  const int* lb = labels + (size_t)b * HW;
  const int* ib = indexes + (size_t)b * HW;

  const int num_tiles = HW / TILE;  // 1024, uniform across block
  for (int t = blockIdx.x; t < num_tiles; t += gridDim.x) {
    const int base = t * TILE;
    // This thread owns pixel (base + tid).
    const int lab = lb[base + tid];
    const int idx = ib[base + tid];
    const int seg = (lab == 255) ? 0 : (lab * (IMAXV + 1) + idx);
    if (seg != 0) atomicAdd(&lcount[seg], 1.0f);

#ifdef USE_TDM
    __syncthreads();  // both stage buffers free (previous tile fully read)
    if (tid < 32) {   // wave 0 drives the DMA pipeline (EXEC all-ones)
      tdm_load_2d((unsigned)(size_t)&stage[0][0], (const void*)(fb + base),
                  TILE, DQ, TILE, HW);
    }
#pragma unroll
    for (int c = 0; c < NCHUNK; ++c) {
      if (tid < 32) {
        if (c + 1 < NCHUNK) {
          // Prefetch next chunk into the other buffer, then wait for the
          // current chunk only (tensor ops of one wave complete in order).
          tdm_load_2d((unsigned)(size_t)&stage[(c + 1) & 1][0],
                      (const void*)(fb + (size_t)((c + 1) * DQ) * HW + base),
                      TILE, DQ, TILE, HW);
          __builtin_amdgcn_s_wait_tensorcnt(1);
        } else {
          __builtin_amdgcn_s_wait_tensorcnt(0);
        }
      }
      __syncthreads();  // chunk c staged, visible to all waves
      if (seg != 0) {
        const float* src = &stage[c & 1][0];
        float* dst = &lsum[seg * LPAD + c * DQ];
#pragma unroll 4
        for (int d = 0; d < DQ; ++d) {
          atomicAdd(&dst[d], src[d * TILE + tid]);  // ds_add_f32 scatter
        }
      }
      __syncthreads();  // chunk c consumed; its buffer may be overwritten
    }
#else
    // Fallback: direct coalesced global loads (contiguous per channel row).
    if (seg != 0) {
      float* dst = &lsum[seg * LPAD];
#pragma unroll 4
      for (int d = 0; d < DIM; ++d) {
        atomicAdd(&dst[d], fb[(size_t)d * HW + base + tid]);
      }
    }
#endif
  }

  __syncthreads();
  // Flush block-local partials to global (skip zeros to cut atomic traffic).
  for (int e = tid; e < NSEG * DIM; e += TILE) {
    const int s = e >> 7;           // / 128
    const int d = e & (DIM - 1);
    const float v = lsum[s * LPAD + d];
    if (v != 0.0f) atomicAdd(&gsum[((size_t)b * NSEG + s) * DIM + d], v);
  }
  if (tid < NSEG) {
    const float c = lcount[tid];
    if (c != 0.0f) atomicAdd(&gcount[b * NSEG + tid], c);
  }
}

// ---------------------------------------------------------------------------
// Phase 2: per-(b,c) instance means, pairwise L1, smooth-L1.
// grid = B*N_CLASSES = 40 blocks, block = 128
// ---------------------------------------------------------------------------
__global__ __launch_bounds__(128) void pair_loss_kernel(
    const float* __restrict__ gsum, const float* __restrict__ gcount,
    float* __restrict__ acc /* [0]=loss sum, [1]=n */) {
  __shared__ float m[(IMAXV + 1) * PPAD];
  __shared__ int inc[IMAXV + 1];
  __shared__ float red;

  const int bc = blockIdx.x;
  const int b = bc / N_CLASSES;
  const int c = bc % N_CLASSES;
  const int tid = threadIdx.x;
  const int seg0 = b * NSEG + c * (IMAXV + 1);

  if (tid == 0) red = 0.0f;
  for (int i = tid; i <= IMAXV; i += 128) {
    const float cnt = gcount[seg0 + i];
    inc[i] = (i > 0 && cnt >= 2.0f) ? 1 : 0;  // skip ignore slot & singletons
  }
  __syncthreads();

  for (int e = tid; e < (IMAXV + 1) * DIM; e += 128) {
    const int i = e >> 7;
    const int d = e & (DIM - 1);
    const float cnt = gcount[seg0 + i];
    const float s = gsum[(size_t)(seg0 + i) * DIM + d];
    m[i * PPAD + d] = inc[i] ? (s / cnt) : 0.0f;
  }
  __syncthreads();

  float part = 0.0f;
  const int npairs = (IMAXV + 1) * (IMAXV + 1);  // 1089 (diagonal contributes 0)
  for (int p = tid; p < npairs; p += 128) {
    const int i = p / (IMAXV + 1);
    const int j = p % (IMAXV + 1);
    if (inc[i] && inc[j]) {
      const float* mi = &m[i * PPAD];
      const float* mj = &m[j * PPAD];
      float s = 0.0f;
#pragma unroll 8
      for (int d = 0; d < DIM; ++d) s += fabsf(mi[d] - mj[d]);
      part += s;
    }
  }
  atomicAdd(&red, part);
  __syncthreads();

  if (tid == 0) {
    int K = 0;
    for (int i = 1; i <= IMAXV; ++i) K += inc[i];
    if (K > 0) {
      const float ret = red / ((float)K * (float)K * (float)DIM);
      const float loss = (ret < 1.0f) ? 0.5f * ret * ret : ret - 0.5f;
      atomicAdd(&acc[0], loss);
      atomicAdd(&acc[1], 1.0f);
    }
  }
}

// ---------------------------------------------------------------------------
// Phase 3: final scalar
// ---------------------------------------------------------------------------
__global__ void finalize_kernel(const float* __restrict__ acc,
                                float* __restrict__ out) {
  const float n = acc[1];
  out[0] = (n > 0.0f) ? (acc[0] / n) / (float)BDIM : 0.0f;
}

// ---------------------------------------------------------------------------
extern "C" void kernel_launch(void* const* d_in, const int* in_sizes, int n_in,
                              void* d_out, int out_size, void* d_ws,
                              size_t ws_size, hipStream_t stream) {
  (void)in_sizes; (void)n_in; (void)out_size; (void)ws_size;
  const float* feat = (const float*)d_in[0];
  const int* labels = (const int*)d_in[1];
  const int* indexes = (const int*)d_in[2];
  float* out = (float*)d_out;

  float* gsum = (float*)d_ws;                       // [B][165][128]
  float* gcount = gsum + (size_t)BDIM * NSEG * DIM; // [B][165]
  float* acc = gcount + (size_t)BDIM * NSEG;        // [2]

  const int nz = BDIM * NSEG * DIM + BDIM * NSEG + 2;
  zero_ws_kernel<<<(nz + 255) / 256, 256, 0, stream>>>((float*)d_ws, nz);

  seg_reduce_kernel<<<dim3(BLOCKS_PER_IMG, BDIM), TILE, 0, stream>>>(
      feat, labels, indexes, gsum, gcount);

  pair_loss_kernel<<<BDIM * N_CLASSES, 128, 0, stream>>>(gsum, gcount, acc);

  finalize_kernel<<<1, 1, 0, stream>>>(acc, out);
}